// CurvphormerLayer_90623809946325
// MI455X (gfx1250) — compile-verified
//
#include <hip/hip_runtime.h>
#include <hip/hip_bf16.h>
#include <float.h>

typedef __attribute__((ext_vector_type(2))) float v2f;
typedef __attribute__((ext_vector_type(8))) float v8f;
typedef __attribute__((ext_vector_type(4))) unsigned int u32x4;
typedef __attribute__((ext_vector_type(8))) int i32x8;
typedef __attribute__((ext_vector_type(4))) int i32x4;

#define N_NODES 50000
#define N_EDGES 800000
#define DIM     128
#define NHEADS  8
#define LN_EPS  1e-5f

// ---------------------------------------------------------------- utilities

__global__ __launch_bounds__(256) void k_fill(float* p, float v, int n) {
  int i = blockIdx.x * blockDim.x + threadIdx.x;
  if (i < n) p[i] = v;
}

__device__ __forceinline__ void atomicMaxF(float* addr, float val) {
  // float max via signed-int max (positives) / unsigned-int min (negatives).
  if (val >= 0.f) atomicMax((int*)addr, __float_as_int(val));
  else            atomicMin((unsigned int*)addr, __float_as_uint(val));
}

// ---------------------------------------------------------------- LayerNorm
// one wave32 per row (D=128 -> 4 floats/lane)
__global__ __launch_bounds__(256) void k_ln(const float* __restrict__ x,
                                            const float* __restrict__ g,
                                            const float* __restrict__ b,
                                            float* __restrict__ out, int rows) {
  int gw   = (blockIdx.x * blockDim.x + threadIdx.x) >> 5;
  int lane = threadIdx.x & 31;
  if (gw >= rows) return;
  const float4 xv = *(const float4*)(x + (size_t)gw * DIM + lane * 4);
  float s = xv.x + xv.y + xv.z + xv.w;
  #pragma unroll
  for (int off = 16; off; off >>= 1) s += __shfl_xor(s, off, 32);
  const float mean = s * (1.0f / DIM);
  float dx = xv.x - mean, dy = xv.y - mean, dz = xv.z - mean, dw = xv.w - mean;
  float vs = dx * dx + dy * dy + dz * dz + dw * dw;
  #pragma unroll
  for (int off = 16; off; off >>= 1) vs += __shfl_xor(vs, off, 32);
  const float inv = rsqrtf(vs * (1.0f / DIM) + LN_EPS);
  const int i0 = lane * 4;
  float4 o;
  o.x = dx * inv * g[i0 + 0] + b[i0 + 0];
  o.y = dy * inv * g[i0 + 1] + b[i0 + 1];
  o.z = dz * inv * g[i0 + 2] + b[i0 + 2];
  o.w = dw * inv * g[i0 + 3] + b[i0 + 3];
  *(float4*)(out + (size_t)gw * DIM + i0) = o;
}

// ---------------------------------------------------------------- WMMA GEMM
// C[M,NC] = A[M,K] @ W[K,NC] + bias[NC] (+ resid[M,NC]) (+ ReLU)
// block = 256 threads = 8 waves; block tile 32x64; wave tile 16x16.
// A block tile (32 x K) is DMA'd into LDS once per block by the Tensor Data
// Mover (tensor_load_to_lds, TENSORcnt) -- the descriptor's tensor_dim1 is
// set to the remaining row count so OOB tail rows are hardware-zeroed.
// Matrix math runs on V_WMMA_F32_16X16X4_F32 (full fp32 precision).
// K and NC are compile-time so all strides fold into instruction offsets;
// only the single partial tail tile (M % 32 != 0) takes the guarded epilogue.
template <int K, int NC, bool RELU, bool RESID>
__global__ __launch_bounds__(256) void k_gemm_f32_wmma(
    const float* __restrict__ A, const float* __restrict__ W,
    const float* __restrict__ bias, const float* __restrict__ resid,
    float* __restrict__ C, int M) {
  __shared__ float Atile[32 * K];

  const int tid  = threadIdx.x;
  const int lane = tid & 31;
  const int wv   = tid >> 5;        // 0..7
  const int m0b  = blockIdx.x * 32;
  const int m0   = m0b + (wv >> 2) * 16;
  const int n0   = blockIdx.y * 64 + (wv & 3) * 16;

  // ---- TDM: stage A[m0b : m0b+32, 0:K] into LDS (wave 0 issues) ----------
  if (tid < 32) {
    const unsigned long long gaddr = (unsigned long long)(A + (size_t)m0b * K);
    const unsigned lds_addr = (unsigned)(size_t)(&Atile[0]);  // low 32 bits = LDS byte addr
    const unsigned rem = (unsigned)(M - m0b);                 // rows left: OOB rows read 0

    u32x4 g0;
    g0[0] = 1u;                                   // count=1, user mode
    g0[1] = lds_addr;                             // lds_addr
    g0[2] = (unsigned)(gaddr & 0xFFFFFFFFu);      // global_addr[31:0]
    g0[3] = (unsigned)((gaddr >> 32) & 0x1FFFFFFu) | 0x80000000u;  // addr[56:32] | type=2

    i32x8 g1;
    g1[0] = (int)(2u << 16);                      // data_size = 4B
    g1[1] = (int)(((unsigned)K & 0xFFFFu) << 16); // tensor_dim0[15:0] (=K)
    g1[2] = (int)(((unsigned)K >> 16) | ((rem & 0xFFFFu) << 16));   // dim0 hi | dim1 lo
    g1[3] = (int)(((rem >> 16) & 0xFFFFu) | ((unsigned)K << 16));   // dim1 hi | tile_dim0=K
    g1[4] = 32;                                   // tile_dim1 = 32 rows
    g1[5] = K;                                    // tensor_dim0_stride = K
    g1[6] = 0;
    g1[7] = 0;

    i32x4 z4 = {0, 0, 0, 0};
    i32x8 z8 = {0, 0, 0, 0, 0, 0, 0, 0};
    __builtin_amdgcn_tensor_load_to_lds(g0, g1, z4, z4, z8, 0);
    __builtin_amdgcn_s_wait_tensorcnt(0);
  }
  __syncthreads();

  const int lm  = lane & 15;        // 0..15
  const int lk2 = (lane >> 4) << 1; // 0 or 2 : K sub-offset per ISA layout
  const float* __restrict__ Ar = &Atile[((wv >> 2) * 16 + lm) * K];
  const int n = n0 + lm;

  v8f acc = {0.f, 0.f, 0.f, 0.f, 0.f, 0.f, 0.f, 0.f};
  #pragma unroll 8
  for (int kk = 0; kk < K; kk += 4) {
    v2f a, bb;
    a.x = Ar[kk + lk2];
    a.y = Ar[kk + lk2 + 1];
    const float* __restrict__ Wp = W + (size_t)(kk + lk2) * NC + n;
    bb.x = Wp[0];
    bb.y = Wp[NC];
    acc = __builtin_amdgcn_wmma_f32_16x16x4_f32(
        /*neg_a=*/false, a, /*neg_b=*/false, bb,
        /*c_mod=*/(short)0, acc, /*reuse_a=*/false, /*reuse_b=*/false);
  }

  const float bn  = bias[n];
  const int   mr0 = m0 + (lane >> 4) * 8;     // first of this lane's 8 rows
  float* __restrict__ Cp = C + (size_t)mr0 * NC + n;
  const float* __restrict__ Rp = RESID ? (resid + (size_t)mr0 * NC + n) : nullptr;

  if (m0b + 32 <= M) {                         // full tile: branch-free stores
    #pragma unroll
    for (int r = 0; r < 8; ++r) {
      float s = acc[r] + bn;
      if (RESID) s += Rp[(size_t)r * NC];
      if (RELU)  s = fmaxf(s, 0.f);
      Cp[(size_t)r * NC] = s;
    }
  } else {                                     // tail tile only
    #pragma unroll
    for (int r = 0; r < 8; ++r) {
      if (mr0 + r < M) {
        float s = acc[r] + bn;
        if (RESID) s += Rp[(size_t)r * NC];
        if (RELU)  s = fmaxf(s, 0.f);
        Cp[(size_t)r * NC] = s;
      }
    }
  }
}

// ---------------------------------------------------------------- edge scores
// one wave32 per edge: curvature projection (all 8 heads) + q.k per head,
// writes scores[E,H] and atomic segment-max keyed by target node.
__global__ __launch_bounds__(256) void k_edge_scores(
    const float* __restrict__ q, const float* __restrict__ k,
    const float* __restrict__ curv, const long long* __restrict__ ei,
    const float* __restrict__ Wc, const float* __restrict__ bc,
    float* __restrict__ scores, float* __restrict__ smax) {
  const int e    = (blockIdx.x * blockDim.x + threadIdx.x) >> 5;
  const int lane = threadIdx.x & 31;
  if (e >= N_EDGES) return;
  const int src = (int)ei[e];
  const int tgt = (int)ei[N_EDGES + e];

  // prefetch the next curvature rows (410MB stream -> keep HBM pipe ahead)
  __builtin_prefetch(curv + (size_t)(e + 16) * DIM + lane * 4, 0, 1);

  // curvature_embeddings[e] @ Wc : each lane covers 4 of 128 channels
  const float4 c = *(const float4*)(curv + (size_t)e * DIM + lane * 4);
  const float cl[4] = {c.x, c.y, c.z, c.w};
  const float* __restrict__ wcp = Wc + (size_t)lane * 4 * NHEADS;
  float p[NHEADS] = {0.f, 0.f, 0.f, 0.f, 0.f, 0.f, 0.f, 0.f};
  #pragma unroll
  for (int j = 0; j < 4; ++j)
    #pragma unroll
    for (int h = 0; h < NHEADS; ++h) p[h] += cl[j] * wcp[j * NHEADS + h];
  #pragma unroll
  for (int h = 0; h < NHEADS; ++h)
    #pragma unroll
    for (int off = 16; off; off >>= 1) p[h] += __shfl_xor(p[h], off, 32);

  // q[src,h] . k[tgt,h] : 4 lanes per head, 4 channels per lane
  const int h2 = lane >> 2, p2 = lane & 3;
  const float4 qv = *(const float4*)(q + (size_t)src * DIM + h2 * 16 + p2 * 4);
  const float4 kv = *(const float4*)(k + (size_t)tgt * DIM + h2 * 16 + p2 * 4);
  float s = qv.x * kv.x + qv.y * kv.y + qv.z * kv.z + qv.w * kv.w;
  s += __shfl_xor(s, 1, 32);
  s += __shfl_xor(s, 2, 32);
  s = s * 0.25f + p[h2] + bc[h2];  // 1/sqrt(HD)=0.25

  if (p2 == 0) {
    scores[(size_t)e * NHEADS + h2] = s;
    atomicMaxF(&smax[(size_t)tgt * NHEADS + h2], s);
  }
}

// exp(score - segmax[tgt]) , accumulate segment sum
__global__ __launch_bounds__(256) void k_expsum(
    float* __restrict__ scores, const long long* __restrict__ ei,
    const float* __restrict__ smax, float* __restrict__ ssum) {
  const int t = blockIdx.x * blockDim.x + threadIdx.x;
  if (t >= N_EDGES * NHEADS) return;
  const int e = t >> 3, h = t & 7;
  const int tgt = (int)ei[N_EDGES + e];
  const float ex = expf(scores[t] - smax[(size_t)tgt * NHEADS + h]);
  scores[t] = ex;
  atomicAdd(&ssum[(size_t)tgt * NHEADS + h], ex);
}

__global__ __launch_bounds__(256) void k_probs(
    float* __restrict__ scores, const long long* __restrict__ ei,
    const float* __restrict__ ssum) {
  const int t = blockIdx.x * blockDim.x + threadIdx.x;
  if (t >= N_EDGES * NHEADS) return;
  const int e = t >> 3, h = t & 7;
  const int tgt = (int)ei[N_EDGES + e];
  scores[t] /= ssum[(size_t)tgt * NHEADS + h];
}

// messages: agg[src] += prob[e,h] * v[tgt] , thread per (edge, channel)
__global__ __launch_bounds__(256) void k_messages(
    const float* __restrict__ probs, const float* __restrict__ v,
    const long long* __restrict__ ei, float* __restrict__ agg) {
  const int t = blockIdx.x * blockDim.x + threadIdx.x;
  if (t >= N_EDGES * DIM) return;
  const int e = t >> 7, d = t & 127, h = d >> 4;
  const int src = (int)ei[e];
  const int tgt = (int)ei[N_EDGES + e];
  const float m = probs[(size_t)e * NHEADS + h] * v[(size_t)tgt * DIM + d];
  atomicAdd(&agg[(size_t)src * DIM + d], m);
}

// ---------------------------------------------------------------- launch
extern "C" void kernel_launch(void* const* d_in, const int* in_sizes, int n_in,
                              void* d_out, int out_size, void* d_ws, size_t ws_size,
                              hipStream_t stream) {
  const float*      x    = (const float*)d_in[0];
  const long long*  ei   = (const long long*)d_in[1];
  const float*      curv = (const float*)d_in[2];
  const float *Wq = (const float*)d_in[3],  *bq = (const float*)d_in[4];
  const float *Wk = (const float*)d_in[5],  *bk = (const float*)d_in[6];
  const float *Wv = (const float*)d_in[7],  *bv = (const float*)d_in[8];
  const float *Wo = (const float*)d_in[9],  *bo = (const float*)d_in[10];
  const float *Wc = (const float*)d_in[11], *bc = (const float*)d_in[12];
  const float *W1 = (const float*)d_in[13], *b1 = (const float*)d_in[14];
  const float *W2 = (const float*)d_in[15], *b2 = (const float*)d_in[16];
  const float *g1 = (const float*)d_in[17], *be1 = (const float*)d_in[18];
  const float *g2 = (const float*)d_in[19], *be2 = (const float*)d_in[20];
  float* out = (float*)d_out;

  const size_t ND = (size_t)N_NODES * DIM;      // 6.4M
  const size_t EH = (size_t)N_EDGES * NHEADS;   // 6.4M
  const size_t NH = (size_t)N_NODES * NHEADS;   // 0.4M
  float* ws     = (float*)d_ws;
  float* xn     = ws;                 // [N,D]   (reused as xn2 later)
  float* q      = ws + ND;            // [N,D]
  float* kk     = ws + 2 * ND;        // [N,D]
  float* v      = ws + 3 * ND;        // [N,D]
  float* sc     = ws + 4 * ND;        // [E,H]  scores -> ex -> probs
  float* smax   = ws + 4 * ND + EH;   // [N,H]
  float* ssum   = smax + NH;          // [N,H]
  float* agg    = ssum + NH;          // [N,D]
  float* hidden = q;                  // [N,4D] reuses dead q..scores region

  // init segment stats + agg
  k_fill<<<(int)((NH + 255) / 256), 256, 0, stream>>>(smax, -FLT_MAX, (int)NH);
  k_fill<<<(int)((NH + 255) / 256), 256, 0, stream>>>(ssum, 0.f, (int)NH);
  k_fill<<<(int)((ND + 255) / 256), 256, 0, stream>>>(agg, 0.f, (int)ND);

  // LN1
  k_ln<<<(N_NODES + 7) / 8, 256, 0, stream>>>(x, g1, be1, xn, N_NODES);

  // Q, K, V projections (TDM-staged WMMA f32)
  dim3 g128((N_NODES + 31) / 32, DIM / 64);
  k_gemm_f32_wmma<128, 128, false, false><<<g128, 256, 0, stream>>>(xn, Wq, bq, nullptr, q,  N_NODES);
  k_gemm_f32_wmma<128, 128, false, false><<<g128, 256, 0, stream>>>(xn, Wk, bk, nullptr, kk, N_NODES);
  k_gemm_f32_wmma<128, 128, false, false><<<g128, 256, 0, stream>>>(xn, Wv, bv, nullptr, v,  N_NODES);

  // edge attention
  k_edge_scores<<<(N_EDGES + 7) / 8, 256, 0, stream>>>(q, kk, curv, ei, Wc, bc, sc, smax);
  k_expsum<<<(int)((EH + 255) / 256), 256, 0, stream>>>(sc, ei, smax, ssum);
  k_probs<<<(int)((EH + 255) / 256), 256, 0, stream>>>(sc, ei, ssum);
  k_messages<<<(int)(((size_t)N_EDGES * DIM + 255) / 256), 256, 0, stream>>>(sc, v, ei, agg);

  // out = agg @ Wo + bo + x   -> x1 (stored in d_out)
  k_gemm_f32_wmma<128, 128, false, true><<<g128, 256, 0, stream>>>(agg, Wo, bo, x, out, N_NODES);

  // LN2 (into xn buffer, now dead)
  k_ln<<<(N_NODES + 7) / 8, 256, 0, stream>>>(out, g2, be2, xn, N_NODES);

  // FFN: hidden = relu(xn2 @ W1 + b1) ; out = x1 + hidden @ W2 + b2
  dim3 g512((N_NODES + 31) / 32, (4 * DIM) / 64);
  k_gemm_f32_wmma<128, 512, true, false><<<g512, 256, 0, stream>>>(xn, W1, b1, nullptr, hidden, N_NODES);
  k_gemm_f32_wmma<512, 128, false, true><<<g128, 256, 0, stream>>>(hidden, W2, b2, out, out, N_NODES);
}